// GetMask_65249143161326
// MI455X (gfx1250) — compile-verified
//
#include <hip/hip_runtime.h>
#include <stdint.h>

#ifndef __has_builtin
#define __has_builtin(x) 0
#endif

#define B_      16
#define H_      1024
#define W_      1024
#define HW_     (H_*W_)            /* 1048576 */
#define HW4_    (HW_/4)            /* 262144  */
#define WROW_   (W_/4)             /* 256 mask words per row */
#define NPLANE_ (B_*3)             /* 48 */
#define NIMG_   ((size_t)NPLANE_*HW_)   /* 50331648 */
#define NW_     (B_*HW_/4)         /* 4194304 mask words */
#define TH_     4                  /* blur output rows per tile */
#define TILESY_ (H_/TH_)           /* 256 */
#define VR_     16                 /* morphology output rows per tile */
#define VROWS_  (VR_+10)           /* 26 LDS rows incl. halo */
#define THRESH_ 0.3f

typedef unsigned int uint32;
typedef unsigned int v4u __attribute__((ext_vector_type(4)));
typedef int          v8i __attribute__((ext_vector_type(8)));
typedef int          v4i __attribute__((ext_vector_type(4)));

#if defined(__HIP_DEVICE_COMPILE__) && __has_builtin(__builtin_amdgcn_tensor_load_to_lds)
#define GM_HAVE_TDM 1
#else
#define GM_HAVE_TDM 0
#endif

#if GM_HAVE_TDM
// TDM 2D tile load: `rows` rows of `row_len` elements (element size 1<<dsz_code
// bytes, dsz_code in {0,1,2,3}) from gsrc (row pitch = row_len elements) into
// LDS at byte offset lds_off, packed. Rows past `valid_rows` are zero-filled by
// the TDM's OOB handling.
__device__ __forceinline__ void tdm_load_rows(const void* gsrc, uint32 lds_off,
                                              uint32 rows, uint32 valid_rows,
                                              uint32 dsz_code, uint32 row_len) {
  unsigned long long ga = (unsigned long long)(uintptr_t)gsrc;
  v4u g0;
  g0.x = 1u;                                                // count=1, user descriptor
  g0.y = lds_off;                                           // LDS byte address
  g0.z = (uint32)(ga & 0xFFFFFFFFull);                      // global_addr[31:0]
  g0.w = (uint32)((ga >> 32) & 0x1FFFFFFull) | (2u << 30);  // global_addr[56:32] | type=2
  v8i g1;
  g1[0] = (int)(dsz_code << 16);                      // data_size
  g1[1] = (int)((row_len & 0xFFFFu) << 16);           // tensor_dim0[15:0] << 16
  g1[2] = (int)((row_len >> 16) |
                ((valid_rows & 0xFFFFu) << 16));      // tensor_dim0 hi | tensor_dim1 lo
  g1[3] = (int)((valid_rows >> 16) |
                ((row_len & 0xFFFFu) << 16));         // tensor_dim1 hi | tile_dim0
  g1[4] = (int)(rows & 0xFFFFu);                      // tile_dim1 | tile_dim2(0)
  g1[5] = (int)row_len;                               // tensor_dim0_stride lo32
  g1[6] = 0;                                          // stride0 hi | stride1 lo (2D: unused)
  g1[7] = 0;
  v4i z4 = {0, 0, 0, 0};
#if defined(__clang_major__) && (__clang_major__ >= 23)
  v8i z8 = {0, 0, 0, 0, 0, 0, 0, 0};
  __builtin_amdgcn_tensor_load_to_lds(g0, g1, z4, z4, z8, 0);
#else
  __builtin_amdgcn_tensor_load_to_lds(g0, g1, z4, z4, 0);
#endif
}
#endif

// -------- K1: 5x5 zero-padded box blur (separable via LDS) + fused reduction --------
// Grid: (TILESY_, NPLANE_, 2).  Writes blurred planes into d_out (used as scratch),
// and per-block {sum, min, max} partials.
__global__ __launch_bounds__(256) void k_blur(const float* __restrict__ nr,
                                              const float* __restrict__ rf,
                                              float* __restrict__ out0,
                                              float* __restrict__ out1,
                                              float4* __restrict__ partials) {
  __shared__ float tile[8 * W_];     // image rows y0-2 .. y0+5 (32 KB)
  __shared__ float vbuf[TH_ * W_];   // vertical 5-tap sums (16 KB, reused for reduce)
  const int tid   = threadIdx.x;
  const int ty    = blockIdx.x;
  const int plane = blockIdx.y;
  const int im    = blockIdx.z;
  const int y0    = ty * TH_;
  const int oy    = (y0 == 0) ? 2 : 0;           // top zero-pad rows
  const int gy    = (y0 == 0) ? 0 : (y0 - 2);    // first global row to fetch
  const int rows  = 8 - oy;                      // rows the DMA writes
  const float* src = (im == 0 ? nr : rf) + (size_t)plane * HW_;
  float*       dst = (im == 0 ? out0 : out1) + (size_t)plane * HW_ + (size_t)y0 * W_;

  if (oy) {  // top edge: zero-pad rows -2,-1
    for (int i = tid; i < 2 * W_; i += 256) tile[i] = 0.f;
  }

#if GM_HAVE_TDM
  if (tid < 32) {  // wave 0 issues the async tensor DMA (EXEC-independent, per-wave)
    uint32 lds_off = (uint32)(uintptr_t)(void*)&tile[oy * W_];
    tdm_load_rows(src + (size_t)gy * W_, lds_off, (uint32)rows, (uint32)(H_ - gy),
                  /*dsz=4B*/2u, (uint32)W_);
    __builtin_amdgcn_s_wait_tensorcnt(0);
  }
#else
  for (int idx = tid; idx < rows * W_; idx += 256) {
    int yy = gy + (idx >> 10);
    tile[oy * W_ + idx] = (yy < H_) ? src[(size_t)yy * W_ + (idx & (W_ - 1))] : 0.f;
  }
#endif
  __syncthreads();

  // Pass 1: vertical 5-tap sums (lane-contiguous columns -> conflict-free LDS)
  for (int idx = tid; idx < TH_ * W_; idx += 256) {
    const float* p = &tile[idx];  // output row r=idx>>10 uses tile rows r..r+4
    vbuf[idx] = p[0] + p[W_] + p[2 * W_] + p[3 * W_] + p[4 * W_];
  }
  __syncthreads();

  // Pass 2: horizontal 5-tap with zero-pad at row ends, store + local reduction
  float tsum = 0.f, tmn = 3.0e38f, tmx = -3.0e38f;
  for (int idx = tid; idx < TH_ * W_; idx += 256) {
    int c = idx & (W_ - 1);
    float h = vbuf[idx];
    if (c >= 1)      h += vbuf[idx - 1];
    if (c >= 2)      h += vbuf[idx - 2];
    if (c <= W_ - 2) h += vbuf[idx + 1];
    if (c <= W_ - 3) h += vbuf[idx + 2];
    float o = h * 0.04f;  // /25
    dst[idx] = o;
    tsum += o; tmn = fminf(tmn, o); tmx = fmaxf(tmx, o);
  }
  __syncthreads();
  float* rs  = vbuf;        // reuse LDS for block reduction
  float* rmn = vbuf + 256;
  float* rmx = vbuf + 512;
  rs[tid] = tsum; rmn[tid] = tmn; rmx[tid] = tmx;
  __syncthreads();
  for (int s = 128; s > 0; s >>= 1) {
    if (tid < s) {
      rs[tid]  += rs[tid + s];
      rmn[tid]  = fminf(rmn[tid], rmn[tid + s]);
      rmx[tid]  = fmaxf(rmx[tid], rmx[tid + s]);
    }
    __syncthreads();
  }
  if (tid == 0) {
    int pidx = (im * NPLANE_ + plane) * TILESY_ + ty;
    partials[pidx] = make_float4(rs[0], rmn[0], rmx[0], 0.f);
  }
}

// -------- K2: final reduction -> brightness-match scalars --------
__global__ __launch_bounds__(256) void k_reduce(const float4* __restrict__ part,
                                                float* __restrict__ sc) {
  __shared__ double dsn[256];
  __shared__ double dsr[256];
  __shared__ float  smn[256];
  __shared__ float  smx[256];
  const int tid = threadIdx.x;
  const int P = NPLANE_ * TILESY_;  // partials per image
  double s0 = 0.0, s1 = 0.0; float mn = 3.0e38f, mx = -3.0e38f;
  for (int i = tid; i < P; i += 256) {
    float4 p = part[i];       // image 0 (non_refer blur)
    s0 += (double)p.x; mn = fminf(mn, p.y); mx = fmaxf(mx, p.z);
    float4 q = part[P + i];   // image 1 (refer blur)
    s1 += (double)q.x;
  }
  dsn[tid] = s0; dsr[tid] = s1; smn[tid] = mn; smx[tid] = mx;
  __syncthreads();
  for (int s = 128; s > 0; s >>= 1) {
    if (tid < s) {
      dsn[tid] += dsn[tid + s];
      dsr[tid] += dsr[tid + s];
      smn[tid]  = fminf(smn[tid], smn[tid + s]);
      smx[tid]  = fmaxf(smx[tid], smx[tid + s]);
    }
    __syncthreads();
  }
  if (tid == 0) {
    float f   = (float)(dsr[0] / dsn[0]);  // mean(r)/mean(nr): counts cancel
    float mnv = smn[0], mxv = smx[0];
    // clip/scale is monotone (f>0) -> global min/max of m from min/max of blur
    float min_m = fminf(fmaxf(mnv * f, 0.f), 1.f);
    float max_m = fminf(fmaxf(mxv * f, 0.f), 1.f);
    float scale = (mxv - mnv) / (max_m - min_m);
    sc[0] = f; sc[1] = min_m; sc[2] = scale; sc[3] = mnv;
  }
}

__device__ __forceinline__ float matchv(float v, float f, float mnm, float scl, float mn) {
  float m = fminf(fmaxf(v * f, 0.f), 1.f);
  return (m - mnm) * scl + mn;
}
__device__ __forceinline__ bool anych(float a0, float a1, float a2,
                                      float b0, float b1, float b2,
                                      float f, float mnm, float scl, float mn) {
  return (fabsf(matchv(a0, f, mnm, scl, mn) - b0) > THRESH_) ||
         (fabsf(matchv(a1, f, mnm, scl, mn) - b1) > THRESH_) ||
         (fabsf(matchv(a2, f, mnm, scl, mn) - b2) > THRESH_);
}

// -------- K3: motion mask, packed 1 byte/pixel (channel-any, broadcast later) --------
__global__ __launch_bounds__(256) void k_mask(const float4* __restrict__ bnr,
                                              const float4* __restrict__ brf,
                                              const float* __restrict__ sc,
                                              uint32* __restrict__ mask) {
  const float f = sc[0], mnm = sc[1], scl = sc[2], mn = sc[3];
  const int stride = gridDim.x * blockDim.x;
  for (int w = blockIdx.x * blockDim.x + threadIdx.x; w < NW_; w += stride) {
    int b = w >> 18;                 // / HW4_
    int rem = w & (HW4_ - 1);
    size_t base = (size_t)b * 3 * HW4_ + rem;
    float4 n0 = bnr[base], n1 = bnr[base + HW4_], n2 = bnr[base + 2 * HW4_];
    float4 r0 = brf[base], r1 = brf[base + HW4_], r2 = brf[base + 2 * HW4_];
    uint32 m = 0;
    m |= anych(n0.x, n1.x, n2.x, r0.x, r1.x, r2.x, f, mnm, scl, mn) ? 0x00000001u : 0u;
    m |= anych(n0.y, n1.y, n2.y, r0.y, r1.y, r2.y, f, mnm, scl, mn) ? 0x00000100u : 0u;
    m |= anych(n0.z, n1.z, n2.z, r0.z, r1.z, r2.z, f, mnm, scl, mn) ? 0x00010000u : 0u;
    m |= anych(n0.w, n1.w, n2.w, r0.w, r1.w, r2.w, f, mnm, scl, mn) ? 0x01000000u : 0u;
    mask[w] = m;
  }
}

// -------- K4: vertical 11-tap erode(min=AND)/dilate(max=OR) on packed bytes ------
// TDM-fed LDS strip: rows [y0-5, y0+VR_+5) of one batch mask plane, u8 elements.
// reduce_window's +/-inf init makes padding neutral -> just skip invalid rows.
template <int IS_MIN>
__global__ __launch_bounds__(256) void k_vpass(const uint32* __restrict__ in,
                                               uint32* __restrict__ out) {
  __shared__ uint32 mtile[VROWS_ * WROW_];   // 26 rows x 1 KB = 26 KB
  const int tid = threadIdx.x;
  const int ty  = blockIdx.x;                // 0 .. H_/VR_-1
  const int b   = blockIdx.y;                // 0 .. B_-1
  const int y0  = ty * VR_;
  const int gy  = (y0 >= 5) ? (y0 - 5) : 0;          // first global row to fetch
  const int oy  = gy - (y0 - 5);                     // top LDS rows left unfetched
  const int yend = (y0 + VR_ + 5 <= H_) ? (y0 + VR_ + 5) : H_;
  const int rows = yend - gy;

#if GM_HAVE_TDM
  if (tid < 32) {
    const uint8_t* gsrc = (const uint8_t*)in + ((size_t)b * H_ + gy) * W_;
    uint32 lds_off = (uint32)(uintptr_t)(void*)&mtile[oy * WROW_];
    tdm_load_rows(gsrc, lds_off, (uint32)rows, (uint32)(H_ - gy),
                  /*dsz=1B*/0u, (uint32)W_);
    __builtin_amdgcn_s_wait_tensorcnt(0);
  }
#else
  for (int idx = tid; idx < rows * WROW_; idx += 256)
    mtile[oy * WROW_ + idx] = in[((size_t)b * H_ + gy) * WROW_ + idx];
#endif
  __syncthreads();

  for (int idx = tid; idx < VR_ * WROW_; idx += 256) {
    int r  = idx >> 8;
    int wx = idx & (WROW_ - 1);
    int y  = y0 + r;
    uint32 acc = IS_MIN ? 0xFFFFFFFFu : 0u;
#pragma unroll
    for (int d = -5; d <= 5; ++d) {
      int yy = y + d;
      if (yy >= 0 && yy < H_) {            // LDS row r+5+d is valid iff yy is
        uint32 v = mtile[(r + 5 + d) * WROW_ + wx];
        acc = IS_MIN ? (acc & v) : (acc | v);
      }
    }
    out[((size_t)b * H_ + y) * WROW_ + wx] = acc;
  }
}

// -------- K4b: horizontal 11-tap erode (byte-wise AND over clamped window) --------
__global__ __launch_bounds__(256) void k_hmin(const uint32* __restrict__ in,
                                              uint32* __restrict__ out) {
  const int stride = gridDim.x * blockDim.x;
  for (int w = blockIdx.x * blockDim.x + threadIdx.x; w < NW_; w += stride) {
    int wr = w & (WROW_ - 1);
    uint32 wd[5];
#pragma unroll
    for (int d = -2; d <= 2; ++d) {
      int r = wr + d;
      wd[d + 2] = (r >= 0 && r < WROW_) ? in[w + d] : 0xFFFFFFFFu;  // neutral
    }
    uint32 res = 0u;
#pragma unroll
    for (int i = 0; i < 4; ++i) {        // byte j of window <-> x = 4*wr + j - 8
      uint32 a = 0xFFu;
#pragma unroll
      for (int j = i + 3; j <= i + 13; ++j)
        a &= (wd[j >> 2] >> ((j & 3) * 8)) & 0xFFu;
      res |= a << (i * 8);
    }
    out[w] = res;
  }
}

// -------- K5: horizontal 11-tap dilate + write ghost / non-ghost (3 channels) ----
__global__ __launch_bounds__(256) void k_final(const uint32* __restrict__ in,
                                               float4* __restrict__ out0,
                                               float4* __restrict__ out1) {
  const int stride = gridDim.x * blockDim.x;
  for (int w = blockIdx.x * blockDim.x + threadIdx.x; w < NW_; w += stride) {
    int wr = w & (WROW_ - 1);
    uint32 wd[5];
#pragma unroll
    for (int d = -2; d <= 2; ++d) {
      int r = wr + d;
      wd[d + 2] = (r >= 0 && r < WROW_) ? in[w + d] : 0u;  // neutral for OR
    }
    float g[4];
#pragma unroll
    for (int i = 0; i < 4; ++i) {
      uint32 a = 0u;
#pragma unroll
      for (int j = i + 3; j <= i + 13; ++j)
        a |= (wd[j >> 2] >> ((j & 3) * 8)) & 0xFFu;
      g[i] = a ? 1.0f : 0.0f;
    }
    float4 gv = make_float4(g[0], g[1], g[2], g[3]);
    float4 ng = make_float4(1.f - g[0], 1.f - g[1], 1.f - g[2], 1.f - g[3]);
    int b = w >> 18;
    int rem = w & (HW4_ - 1);
    size_t base = (size_t)b * 3 * HW4_ + rem;
    out0[base] = gv; out0[base + HW4_] = gv; out0[base + 2 * HW4_] = gv;
    out1[base] = ng; out1[base + HW4_] = ng; out1[base + 2 * HW4_] = ng;
  }
}

extern "C" void kernel_launch(void* const* d_in, const int* in_sizes, int n_in,
                              void* d_out, int out_size, void* d_ws, size_t ws_size,
                              hipStream_t stream) {
  const float* nr = (const float*)d_in[0];
  const float* rf = (const float*)d_in[1];
  float* out0 = (float*)d_out;           // ghost_mask; also blur(nr) scratch
  float* out1 = out0 + NIMG_;            // non_ghost_mask; also blur(rf) scratch

  // workspace layout: [0,16)B scalars | [256, 256+24576*16) partials | masks
  float*  sc       = (float*)d_ws;
  float4* partials = (float4*)((char*)d_ws + 256);
  uint32* maskA    = (uint32*)((char*)d_ws + 512 * 1024);
  uint32* maskB    = maskA + NW_;

  dim3 gb(TILESY_, NPLANE_, 2);
  k_blur<<<gb, 256, 0, stream>>>(nr, rf, out0, out1, partials);
  k_reduce<<<1, 256, 0, stream>>>(partials, sc);

  const int blk = 8192;
  k_mask<<<blk, 256, 0, stream>>>((const float4*)out0, (const float4*)out1, sc, maskA);

  dim3 gv(H_ / VR_, B_);
  k_vpass<1><<<gv, 256, 0, stream>>>(maskA, maskB);   // erode: vertical min (AND)
  k_hmin<<<blk, 256, 0, stream>>>(maskB, maskA);      // erode: horizontal min
  k_vpass<0><<<gv, 256, 0, stream>>>(maskA, maskB);   // dilate: vertical max (OR)
  k_final<<<blk, 256, 0, stream>>>(maskB, (float4*)out0, (float4*)out1);

  (void)in_sizes; (void)n_in; (void)out_size; (void)ws_size;
}